// ValidConvolution_53764400611580
// MI455X (gfx1250) — compile-verified
//
#include <hip/hip_runtime.h>

typedef __attribute__((ext_vector_type(2))) float v2f;
typedef __attribute__((ext_vector_type(8))) float v8f;

#define N_SITES_C 262144
#define F_VOL_C 27
#define R_PER_F_C 65536
#define N_IN_C 32
#define N_OUT_C 32

#define RULES_PER_BLOCK 1024
#define BLOCKS_PER_F (R_PER_F_C / RULES_PER_BLOCK)   // 64
#define TILES_PER_BLOCK (RULES_PER_BLOCK / 16)       // 64
#define WAVES_PER_BLOCK 8

// ---------------------------------------------------------------------------
// out[n, o] = bias[o] for all sites (float4 writes; 8M floats total)
// ---------------------------------------------------------------------------
__global__ __launch_bounds__(256)
void bias_init_kernel(const float* __restrict__ bias, float* __restrict__ out) {
    const int idx = blockIdx.x * blockDim.x + threadIdx.x;   // one float4 each
    const float4* b4 = (const float4*)bias;                  // 32 floats = 8 float4
    ((float4*)out)[idx] = b4[idx & 7];                       // (idx*4)%32 / 4
}

// ---------------------------------------------------------------------------
// Rulebook sparse conv: for offset f, rule r:
//   out[rules_out[f,r]] += features[rules_in[f,r]] @ W[f]   (W[f]: 32x32)
// One wave processes 16 rules/tile with V_WMMA_F32_16X16X4_F32.
// ---------------------------------------------------------------------------
__global__ __launch_bounds__(256)
void valid_conv_wmma_kernel(const float* __restrict__ features,
                            const float* __restrict__ weight,
                            const int*   __restrict__ rules_in,
                            const int*   __restrict__ rules_out,
                            float*       __restrict__ out) {
    const int lane  = threadIdx.x & 31;
    const int wave  = threadIdx.x >> 5;
    const int f     = blockIdx.x / BLOCKS_PER_F;
    const int chunk = blockIdx.x % BLOCKS_PER_F;
    const int base0 = chunk * RULES_PER_BLOCK;

    const float* __restrict__ Wf   = weight    + (size_t)f * N_IN_C * N_OUT_C;
    const int*   __restrict__ rin  = rules_in  + (size_t)f * R_PER_F_C;
    const int*   __restrict__ rout = rules_out + (size_t)f * R_PER_F_C;

    const int half = lane >> 4;   // 0: K pair {0,1} / rows M=j ; 1: K pair {2,3} / rows M=j+8
    const int lid  = lane & 15;   // column / row-index lane

    // ---- B fragments for W[f] (loaded once per block; 4KB block, L2-resident)
    // B(4x16) wave32 layout: lanes 0-15 hold (K=k0,k0+1, N=lid), lanes 16-31 hold (K=k0+2,k0+3, N=lid)
    v2f bfrag[8][2];
#pragma unroll
    for (int kb = 0; kb < 8; ++kb) {
        const int klo = kb * 4 + half * 2;
#pragma unroll
        for (int nb = 0; nb < 2; ++nb) {
            const int n = lid + nb * 16;
            v2f b;
            b.x = Wf[(klo + 0) * N_OUT_C + n];
            b.y = Wf[(klo + 1) * N_OUT_C + n];
            bfrag[kb][nb] = b;
        }
    }

    for (int t = wave; t < TILES_PER_BLOCK; t += WAVES_PER_BLOCK) {
        const int base = base0 + t * 16;
        const int ri = rin[base + lid];   // gather row index for row (lane&15)
        const int ro = rout[base + lid];  // scatter row index for row (lane&15)

        // ---- A fragments: 16 rows x K=32, each lane holds a float2 slice per K-block
        const float* frow = features + (size_t)ri * N_IN_C + half * 2;
        v2f afrag[8];
#pragma unroll
        for (int kb = 0; kb < 8; ++kb)
            afrag[kb] = *(const v2f*)(frow + kb * 4);   // global_load_b64, L2 hits

        // ---- 16x32 += (16x32) @ (32x32): 8 K-steps x 2 N-tiles of f32 WMMA
        v8f acc0 = {};
        v8f acc1 = {};
#pragma unroll
        for (int kb = 0; kb < 8; ++kb) {
            acc0 = __builtin_amdgcn_wmma_f32_16x16x4_f32(
                false, afrag[kb], false, bfrag[kb][0], (short)0, acc0, false, false);
            acc1 = __builtin_amdgcn_wmma_f32_16x16x4_f32(
                false, afrag[kb], false, bfrag[kb][1], (short)0, acc1, false, false);
        }

        // ---- Scatter-accumulate: D layout -> lane lid = column, VGPR j = row (+8 for half 1)
#pragma unroll
        for (int j = 0; j < 8; ++j) {
            const int row  = j + half * 8;
            const int oidx = __shfl(ro, row, 32);
            float* dst = out + (size_t)oidx * N_OUT_C + lid;
            atomicAdd(dst,      acc0[j]);   // global_atomic_add_f32 (no return)
            atomicAdd(dst + 16, acc1[j]);
        }
    }
}

// ---------------------------------------------------------------------------
extern "C" void kernel_launch(void* const* d_in, const int* in_sizes, int n_in,
                              void* d_out, int out_size, void* d_ws, size_t ws_size,
                              hipStream_t stream) {
    const float* features  = (const float*)d_in[0];  // [262144, 32] f32
    const float* weight    = (const float*)d_in[1];  // [864, 32] f32
    const float* bias      = (const float*)d_in[2];  // [32] f32
    const int*   rules_in  = (const int*)  d_in[3];  // [27, 65536] i32
    const int*   rules_out = (const int*)  d_in[4];  // [27, 65536] i32
    float* out = (float*)d_out;                      // [262144, 32] f32

    (void)in_sizes; (void)n_in; (void)out_size; (void)d_ws; (void)ws_size;

    // 1) out = broadcast(bias)
    const int vec4_total = N_SITES_C * N_OUT_C / 4;  // 2M float4
    bias_init_kernel<<<vec4_total / 256, 256, 0, stream>>>(bias, out);

    // 2) scatter-accumulated WMMA GEMMs (same stream -> ordered after init)
    valid_conv_wmma_kernel<<<F_VOL_C * BLOCKS_PER_F, 256, 0, stream>>>(
        features, weight, rules_in, rules_out, out);
}